// FocalLossAdaptive_89670327205969
// MI455X (gfx1250) — compile-verified
//
#include <hip/hip_runtime.h>
#include <cfloat>

// Focal loss with adaptive gamma, N x C fp32 logits, int targets.
// Single-pass online log-sum-exp per row (base-2 scaled domain);
// HBM-bandwidth-bound streaming kernel with 4-way MLP unroll.

typedef float v4f __attribute__((ext_vector_type(4)));

#define L2E 1.44269504088896340736f   // log2(e)
#define LN2 0.69314718055994530942f   // ln(2)
#define BLK 320                       // 10 wave32 waves; 32000/4/320 = 25 loads/thread

__device__ __forceinline__ float fexp2(float x) {
    return __builtin_amdgcn_exp2f(x);  // v_exp_f32 (base-2 native)
}

// Online update of (ml, s) with one float4, all in base-2 scaled domain:
//   ml tracks max(x)*log2e,  s = sum 2^(x*log2e - ml).
// 2x v_max3 + 4x v_mul(prescale) + 5x v_sub + 5x v_exp + fma/add chain.
__device__ __forceinline__ void upd(float& ml, float& s, v4f v) {
    v4f xl = v * L2E;
    float mn = fmaxf(fmaxf(fmaxf(xl.x, xl.y), xl.z), fmaxf(xl.w, ml));
    s = s * fexp2(ml - mn)
      + fexp2(xl.x - mn) + fexp2(xl.y - mn)
      + fexp2(xl.z - mn) + fexp2(xl.w - mn);
    ml = mn;
}

// Combine two partials. Identity = (-FLT_MAX, 0): identity-vs-identity gives
// exp2(0)=1 -> s=0; identity-vs-real gives exp2(~-3.4e38)=0 -> 0*0=0. No NaN.
__device__ __forceinline__ void comb(float& ml, float& s, float ml2, float s2) {
    float mn = fmaxf(ml, ml2);
    s = s * fexp2(ml - mn) + s2 * fexp2(ml2 - mn);
    ml = mn;
}

__global__ __launch_bounds__(BLK) void focal_row_kernel(
        const float* __restrict__ inp,       // [N, C]
        const int*   __restrict__ tgt,       // [N]
        float*       __restrict__ row_loss,  // [N] (workspace)
        int C) {
    const int row = blockIdx.x;
    const int tid = threadIdx.x;
    const float* __restrict__ rowp = inp + (size_t)row * (size_t)C;

    // Target logit: fetched once, early, by the finalizing thread.
    float xt = 0.0f;
    if (tid == 0) xt = rowp[tgt[row]];

    const v4f* __restrict__ p4 = (const v4f*)rowp;
    const int Q = C >> 2;  // float4 count per row

    // 4 independent accumulators -> 4 outstanding b128 NT loads per wave iter.
    float ml0 = -FLT_MAX, ml1 = -FLT_MAX, ml2 = -FLT_MAX, ml3 = -FLT_MAX;
    float s0 = 0.f, s1 = 0.f, s2 = 0.f, s3 = 0.f;

    int i = tid;
    for (; i + 3 * BLK < Q; i += 4 * BLK) {
        v4f a = __builtin_nontemporal_load(p4 + i);
        v4f b = __builtin_nontemporal_load(p4 + i + BLK);
        v4f c = __builtin_nontemporal_load(p4 + i + 2 * BLK);
        v4f d = __builtin_nontemporal_load(p4 + i + 3 * BLK);
        if (i + 16 * BLK < Q)  // gfx1250 global_prefetch_b8, ~80KB ahead
            __builtin_prefetch((const void*)(p4 + i + 16 * BLK), 0, 0);
        upd(ml0, s0, a);
        upd(ml1, s1, b);
        upd(ml2, s2, c);
        upd(ml3, s3, d);
    }
    for (; i < Q; i += BLK) {
        v4f a = __builtin_nontemporal_load(p4 + i);
        upd(ml0, s0, a);
    }
    comb(ml0, s0, ml1, s1);
    comb(ml2, s2, ml3, s3);
    comb(ml0, s0, ml2, s2);

    // scalar tail (C % 4 != 0); no-op for C = 32000
    for (int j = (Q << 2) + tid; j < C; j += BLK) {
        float xl = __builtin_nontemporal_load(rowp + j) * L2E;
        float mn = fmaxf(ml0, xl);
        s0 = s0 * fexp2(ml0 - mn) + fexp2(xl - mn);
        ml0 = mn;
    }

    // ---- wave32 shuffle reduction ----
    for (int off = 16; off > 0; off >>= 1) {
        float m2 = __shfl_xor(ml0, off, 32);
        float t2 = __shfl_xor(s0, off, 32);
        comb(ml0, s0, m2, t2);
    }

    __shared__ float smm[16];
    __shared__ float sms[16];
    const int wave = tid >> 5;
    const int lane = tid & 31;
    if (lane == 0) { smm[wave] = ml0; sms[wave] = s0; }
    __syncthreads();

    if (wave == 0) {
        const int nw = BLK >> 5;  // 10 waves
        float mm = (lane < nw) ? smm[lane] : -FLT_MAX;
        float ss = (lane < nw) ? sms[lane] : 0.0f;
        for (int off = 16; off > 0; off >>= 1) {
            float m2 = __shfl_xor(mm, off, 32);
            float t2 = __shfl_xor(ss, off, 32);
            comb(mm, ss, m2, t2);
        }
        if (lane == 0) {
            // ln(sum e^(x - max)) + max = LN2 * (mm + log2(ss))   [scaled domain]
            float lse   = LN2 * (mm + log2f(ss));
            float logpt = xt - lse;                 // log_softmax at target
            float pt    = fexp2(logpt * L2E);       // e^logpt
            // gamma buckets: pt>=0.5 -> 3, pt<0.2 -> 5, else 3  (exact int powers)
            float u  = 1.0f - pt;
            float u3 = u * u * u;
            float w  = (pt < 0.2f) ? (u3 * u * u) : u3;
            row_loss[row] = -w * logpt;
        }
    }
}

// Deterministic mean of N row losses (single block). For the N==4096 fast path
// the 16KB of partials are staged into LDS with gfx1250 async-to-LDS copies
// (ASYNCcnt path, ISA 15.18.3 op 98), then tree-reduced from LDS.
__global__ __launch_bounds__(256) void reduce_mean_kernel(
        const float* __restrict__ x, float* __restrict__ out, int n) {
    __shared__ float sbuf[4096];   // 16 KB staging
    __shared__ float sm[256];

    float a = 0.0f;
    if (n == 4096) {
        const unsigned lbase = (unsigned)(uintptr_t)(void*)&sbuf[0];
        const unsigned long long gbase = (unsigned long long)(uintptr_t)x;
        #pragma unroll
        for (int k = 0; k < 4; ++k) {
            unsigned goff  = ((unsigned)threadIdx.x + k * 256u) * 16u;  // bytes
            unsigned laddr = lbase + goff;
            // GVS: per-lane LDS dest VGPR, per-lane 32-bit offset VGPR, SGPR64 base
            asm volatile("global_load_async_to_lds_b128 %0, %1, %2"
                         :: "v"(laddr), "v"(goff), "s"(gbase) : "memory");
        }
        asm volatile("s_wait_asynccnt 0x0" ::: "memory");
        __syncthreads();
        #pragma unroll
        for (int k = 0; k < 16; ++k) a += sbuf[threadIdx.x + k * 256];
    } else {
        for (int i = threadIdx.x; i < n; i += 256) a += x[i];
    }

    sm[threadIdx.x] = a;
    __syncthreads();
    for (int sft = 128; sft > 0; sft >>= 1) {
        if ((int)threadIdx.x < sft) sm[threadIdx.x] += sm[threadIdx.x + sft];
        __syncthreads();
    }
    if (threadIdx.x == 0) out[0] = sm[0] / (float)n;
}

extern "C" void kernel_launch(void* const* d_in, const int* in_sizes, int n_in,
                              void* d_out, int out_size, void* d_ws, size_t ws_size,
                              hipStream_t stream) {
    const float* inp = (const float*)d_in[0];
    const int*   tgt = (const int*)d_in[1];

    const int N = in_sizes[1];                 // 4096
    const int C = (int)(in_sizes[0] / N);      // 32000

    float* row_loss = (float*)d_ws;            // N floats of scratch
    float* out      = (float*)d_out;           // scalar fp32 mean

    focal_row_kernel<<<dim3(N), dim3(BLK), 0, stream>>>(inp, tgt, row_loss, C);
    reduce_mean_kernel<<<dim3(1), dim3(256), 0, stream>>>(row_loss, out, N);
}